// SimpleConvNet_11287174053918
// MI455X (gfx1250) — compile-verified
//
#include <hip/hip_runtime.h>

// ---------------------------------------------------------------------------
// Fused sparse conv3x3 -> leaky(0.5) -> conv3x3 (masked) on MI455X (gfx1250).
// Memory-bound (~3 FLOP/byte): one fused pass keeps the a1 intermediate in LDS
// (saves ~270 MB of HBM/L2 traffic vs two passes). fp32 WMMA
// (v_wmma_f32_16x16x4_f32) does both convs as implicit GEMM at full reference
// precision: Y[16px x 16co] = A[16 x 28] * W[28 x 16] in 7 K=4 steps.
// Interior tiles are staged with global_load_async_to_lds_b128 (ASYNCcnt).
// ---------------------------------------------------------------------------

typedef float v2f __attribute__((ext_vector_type(2)));
typedef float v8f __attribute__((ext_vector_type(8)));

#define BD 8
#define HD 1024
#define WD 1024
#define SITES (BD * HD * WD)   // 8,388,608 sites
#define TY 8                   // output rows per block
#define TX 126                 // output cols per block
#define C1W 128                // conv1 strip cols = TX+2 = 8 waves * 16 px
#define DLW 130                // dense/a1 tile stride (TX+4)
#define DROWS 12               // dense tile rows (TY+4)
#define A1ROWS 10              // a1 tile rows (TY+2)
#define NEG_SLOPE 0.5f

// im2col K index -> float offset inside a stride-DLW float4 tile (as float*),
// relative to the center pixel's float4 base. K = (ky*3+kx)*3 + ci.
constexpr int koff(int k) {
    int tap = k / 3, ch = k % 3;
    return ((tap / 3 - 1) * DLW + (tap % 3 - 1)) * 4 + ch;
}

// A-fragment element, ISA 16x4 fp32 layout:
//   a.x = A[M=lane&15][4*CC + 2*hi + 0], a.y = ... + 1   (hi = lane>>4)
template <int CC, int J>
__device__ __forceinline__ float aval(const float* tf, int base, int hi) {
    constexpr int k0 = 4 * CC + J;       // hi == 0 (max 25, valid)
    constexpr int k1 = 4 * CC + 2 + J;   // hi == 1 (27 only at CC=6,J=1)
    constexpr int o0 = koff(k0);
    constexpr int o1 = koff(k1 < 27 ? k1 : 0);
    float v = tf[base + (hi ? o1 : o0)];
    if (k1 >= 27 && hi) v = 0.f;         // zero-pad K=27
    return v;
}

__global__ __launch_bounds__(256) void zero4_kernel(float4* __restrict__ p, int n) {
    int i = blockIdx.x * 256 + threadIdx.x;
    if (i < n) p[i] = make_float4(0.f, 0.f, 0.f, 0.f);
}

__global__ __launch_bounds__(256) void scatter_kernel(
    const float* __restrict__ f, const int* __restrict__ ib,
    const int* __restrict__ iy, const int* __restrict__ ix,
    float4* __restrict__ dense, int n) {
    int i = blockIdx.x * 256 + threadIdx.x;
    if (i < n) {
        size_t site = ((size_t)ib[i] * HD + iy[i]) * WD + ix[i];
        dense[site] = make_float4(f[3 * i], f[3 * i + 1], f[3 * i + 2], 1.f);
    }
}

__global__ __launch_bounds__(256, 2) void fused_conv_kernel(
    const float4* __restrict__ in4,
    const float* __restrict__ w1f, const float* __restrict__ b1,
    const float* __restrict__ w2f, const float* __restrict__ b2,
    float* __restrict__ out3) {
    __shared__ float4 dt[DROWS * DLW];         // dense tile: rows y0-2..y0+9
    __shared__ float4 a1t[A1ROWS * DLW + 2];   // a1 tile: rows y0-1..y0+8 (+pad)
    __shared__ float ldso[8 * 16 * 3];         // per-wave D-transpose scratch

    const int tid = threadIdx.x;
    const int wv = tid >> 5, lane = tid & 31, lo = lane & 15, hi = lane >> 4;
    const int y0 = blockIdx.y * TY;
    const int x0 = blockIdx.x * TX;
    const size_t plane = (size_t)blockIdx.z * HD * WD;

    // ---- stage dense tile (cols x0-2..x0+127, rows y0-2..y0+9) ----
    const bool interior = (x0 >= 2) && (x0 + DLW - 2 <= WD) &&
                          (y0 >= 2) && (y0 + DROWS - 2 <= HD);
    if (interior) {
        // gfx1250 async copy global->LDS, tracked by ASYNCcnt
        const float4* src = in4 + plane + (size_t)(y0 - 2) * WD + (x0 - 2);
        for (int idx = tid; idx < DROWS * DLW; idx += 256) {
            int r = idx / DLW, col = idx - r * DLW;
            unsigned lds = (unsigned)(uintptr_t)(void*)&dt[idx];
            unsigned long long ga =
                (unsigned long long)(uintptr_t)(const void*)(src + (size_t)r * WD + col);
            asm volatile("global_load_async_to_lds_b128 %0, %1, off"
                         :: "v"(lds), "v"(ga) : "memory");
        }
        asm volatile("s_wait_asynccnt 0" ::: "memory");
    } else {
        for (int idx = tid; idx < DROWS * DLW; idx += 256) {
            int r = idx / DLW, col = idx - r * DLW;
            int gy = y0 + r - 2, gx = x0 + col - 2;
            float4 v = make_float4(0.f, 0.f, 0.f, 0.f);
            if (gy >= 0 && gy < HD && gx >= 0 && gx < WD)
                v = in4[plane + (size_t)gy * WD + gx];
            dt[idx] = v;
        }
    }
    __syncthreads();

    // ---- B (weight) fragments for both convs: B[K][co], K->28, co->16 pad.
    // ISA 4x16 fp32 B layout: b.x = B[2*hi][lo], b.y = B[2*hi+1][lo].
    const int n = lo;
    v2f w1w0, w1w1, w1w2, w1w3, w1w4, w1w5, w1w6;
    v2f w2w0, w2w1, w2w2, w2w3, w2w4, w2w5, w2w6;
#define BLDW(WP, CC, VAR)                                         \
    {                                                             \
        int ka = 4 * CC + 2 * hi;                                 \
        int kb = ka + 1;                                          \
        VAR.x = (n < 3) ? WP[ka * 3 + n] : 0.f;                   \
        VAR.y = (n < 3 && kb < 27) ? WP[kb * 3 + n] : 0.f;        \
    }
    BLDW(w1f, 0, w1w0) BLDW(w1f, 1, w1w1) BLDW(w1f, 2, w1w2) BLDW(w1f, 3, w1w3)
    BLDW(w1f, 4, w1w4) BLDW(w1f, 5, w1w5) BLDW(w1f, 6, w1w6)
    BLDW(w2f, 0, w2w0) BLDW(w2f, 1, w2w1) BLDW(w2f, 2, w2w2) BLDW(w2f, 3, w2w3)
    BLDW(w2f, 4, w2w4) BLDW(w2f, 5, w2w5) BLDW(w2f, 6, w2w6)
#undef BLDW
    const float bs10 = b1[0], bs11 = b1[1], bs12 = b1[2];
    const float bs20 = b2[0], bs21 = b2[1], bs22 = b2[2];

#define STEP(TF, CC, BVAR)                                                     \
    {                                                                          \
        v2f a;                                                                 \
        a.x = aval<CC, 0>(TF, base, hi);                                       \
        a.y = aval<CC, 1>(TF, base, hi);                                       \
        c = __builtin_amdgcn_wmma_f32_16x16x4_f32(false, a, false, BVAR,       \
                                                  (short)0, c, false, false);  \
    }

    // ================= phase 1: conv1 + bias + mask + leaky -> a1 (LDS) ====
    // a1 site (rr, c1): gy = y0-1+rr, gx = x0-1+c1; dense center (rr+1, c1+1)
    const float* tf = (const float*)dt;
    const int cx = wv * 16 + lo;   // strip col 0..127
    for (int rr = 0; rr < A1ROWS; ++rr) {
        const int base = ((rr + 1) * DLW + (cx + 1)) * 4;
        v8f c = {0.f, 0.f, 0.f, 0.f, 0.f, 0.f, 0.f, 0.f};
        STEP(tf, 0, w1w0) STEP(tf, 1, w1w1) STEP(tf, 2, w1w2) STEP(tf, 3, w1w3)
        STEP(tf, 4, w1w4) STEP(tf, 5, w1w5) STEP(tf, 6, w1w6)

        if (lo < 3) {   // D[M=r+8*hi][N=lo] -> pixel-major scratch
#pragma unroll
            for (int r = 0; r < 8; ++r)
                ldso[(wv * 16 + (r + 8 * hi)) * 3 + lo] = c[r];
        }
        if (lane < 16) {
            float ms = 0.f;   // 3x3 dilation of input mask (.w plane, all >= 0)
#pragma unroll
            for (int dy = -1; dy <= 1; ++dy)
#pragma unroll
                for (int dx = -1; dx <= 1; ++dx)
                    ms += tf[base + (dy * DLW + dx) * 4 + 3];
            const int gy = y0 - 1 + rr, gx = x0 - 1 + cx;
            const bool act = (ms > 0.f) &&
                             (gy >= 0 && gy < HD && gx >= 0 && gx < WD);
            float v0 = ldso[(wv * 16 + lo) * 3 + 0] + bs10;
            float v1 = ldso[(wv * 16 + lo) * 3 + 1] + bs11;
            float v2 = ldso[(wv * 16 + lo) * 3 + 2] + bs12;
            if (!act) { v0 = 0.f; v1 = 0.f; v2 = 0.f; }
            v0 = v0 > 0.f ? v0 : NEG_SLOPE * v0;
            v1 = v1 > 0.f ? v1 : NEG_SLOPE * v1;
            v2 = v2 > 0.f ? v2 : NEG_SLOPE * v2;
            a1t[rr * DLW + (cx + 1)] = make_float4(v0, v1, v2, act ? 1.f : 0.f);
        }
    }
    __syncthreads();   // a1 tile read across wave strips in phase 2

    // ================= phase 2: conv2 + bias + mask -> out ================
    // out site (rr, c2): gy = y0+rr, gx = x0+c2; a1 center (rr+1, c2+2).
    // cols c2 in [126,128) are strip padding: computed, discarded at store.
    const float* af = (const float*)a1t;
    for (int rr = 0; rr < TY; ++rr) {
        const int base = ((rr + 1) * DLW + (cx + 2)) * 4;
        v8f c = {0.f, 0.f, 0.f, 0.f, 0.f, 0.f, 0.f, 0.f};
        STEP(af, 0, w2w0) STEP(af, 1, w2w1) STEP(af, 2, w2w2) STEP(af, 3, w2w3)
        STEP(af, 4, w2w4) STEP(af, 5, w2w5) STEP(af, 6, w2w6)

        if (lo < 3) {
#pragma unroll
            for (int r = 0; r < 8; ++r)
                ldso[(wv * 16 + (r + 8 * hi)) * 3 + lo] = c[r];
        }
        if (lane < 16) {
            float ms = 0.f;   // 3x3 dilation of a1 mask
#pragma unroll
            for (int dy = -1; dy <= 1; ++dy)
#pragma unroll
                for (int dx = -1; dx <= 1; ++dx)
                    ms += af[base + (dy * DLW + dx) * 4 + 3];
            const bool act = ms > 0.f;
            float v0 = ldso[(wv * 16 + lo) * 3 + 0] + bs20;
            float v1 = ldso[(wv * 16 + lo) * 3 + 1] + bs21;
            float v2 = ldso[(wv * 16 + lo) * 3 + 2] + bs22;
            if (!act) { v0 = 0.f; v1 = 0.f; v2 = 0.f; }
            const int gy = y0 + rr, gx = x0 + cx;
            if (cx < TX && gx < WD) {
                const size_t site = plane + (size_t)gy * WD + gx;
                out3[site * 3 + 0] = v0;
                out3[site * 3 + 1] = v1;
                out3[site * 3 + 2] = v2;
            }
        }
    }
#undef STEP
}

extern "C" void kernel_launch(void* const* d_in, const int* in_sizes, int n_in,
                              void* d_out, int out_size, void* d_ws, size_t ws_size,
                              hipStream_t stream) {
    const float* feats = (const float*)d_in[0];
    const int* ib = (const int*)d_in[1];
    const int* iy = (const int*)d_in[2];
    const int* ix = (const int*)d_in[3];
    const float* w1 = (const float*)d_in[4];
    const float* b1 = (const float*)d_in[5];
    const float* w2 = (const float*)d_in[6];
    const float* b2 = (const float*)d_in[7];
    const int n = in_sizes[1];   // 1.5M points

    float4* dense = (float4*)d_ws;   // 128 MiB dense grid (xyz=feat, w=mask)

    zero4_kernel<<<SITES / 256, 256, 0, stream>>>(dense, SITES);
    scatter_kernel<<<(n + 255) / 256, 256, 0, stream>>>(feats, ib, iy, ix, dense, n);

    dim3 grid((WD + TX - 1) / TX, HD / TY, BD);   // 9 x 128 x 8 blocks
    fused_conv_kernel<<<grid, 256, 0, stream>>>(dense, w1, b1, w2, b2,
                                                (float*)d_out);
}